// SoftSymmetricAlignment_23227183137333
// MI455X (gfx1250) — compile-verified
//
#include <hip/hip_runtime.h>
#include <hip/hip_bf16.h>
#include <stdint.h>

// Problem constants (fixed by the reference setup_inputs()).
#define BB 4
#define NN 512
#define MM 512
#define DD 128

typedef __attribute__((ext_vector_type(4))) unsigned int u32x4_t;
typedef __attribute__((ext_vector_type(8))) int          i32x8_t;
typedef __attribute__((ext_vector_type(4))) int          i32x4_t;

// Low 32 bits of a generic pointer to LDS == LDS byte offset (aperture scheme).
__device__ __forceinline__ uint32_t lds_offset(const void* p) {
    return (uint32_t)(uintptr_t)p;
}

// Issue a TDM 2D tile load (f32 elements) into LDS.
// D# layout per CDNA5 ISA:
//  group0: [1:0]=count=1, [63:32]=lds_addr, [120:64]=global_addr, [127:126]=type=2
//  group1: [17:16]=data_size(2 -> 4B), [79:48]=tensor_dim0, [111:80]=tensor_dim1,
//          [127:112]=tile_dim0, [143:128]=tile_dim1, [207:160]=tensor_dim0_stride
__device__ __forceinline__ void tdm_load_tile_f32(uint32_t lds_addr,
                                                  const float* gptr,
                                                  uint32_t width_elems,
                                                  uint32_t rows,
                                                  uint32_t row_stride_elems) {
    const uint64_t ga = (uint64_t)(uintptr_t)gptr;
    u32x4_t g0;
    g0[0] = 1u;                                                 // count=1, user desc
    g0[1] = lds_addr;                                           // lds_addr (bytes)
    g0[2] = (uint32_t)ga;                                       // global_addr[31:0]
    g0[3] = ((uint32_t)(ga >> 32) & 0x01FFFFFFu) | (2u << 30);  // addr[56:32] | type=2
    i32x8_t g1;
    g1[0] = (int)(2u << 16);                                    // data_size=4B, mask=0
    g1[1] = (int)((width_elems & 0xFFFFu) << 16);               // tensor_dim0 lo16
    g1[2] = (int)(((width_elems >> 16) & 0xFFFFu)
                  | ((rows & 0xFFFFu) << 16));                  // td0 hi16 | td1 lo16
    g1[3] = (int)(((rows >> 16) & 0xFFFFu)
                  | ((width_elems & 0xFFFFu) << 16));           // td1 hi16 | tile_dim0
    g1[4] = (int)(rows & 0xFFFFu);                              // tile_dim1 (tile_dim2=0)
    g1[5] = (int)row_stride_elems;                              // tensor_dim0_stride lo32
    g1[6] = 0;                                                  // stride hi bits = 0
    g1[7] = 0;
    i32x4_t gz4 = {0, 0, 0, 0};                                 // groups 2/3 unused (2D)
    i32x8_t gz8 = {0, 0, 0, 0, 0, 0, 0, 0};                     // extra group (clang-23 form)
    __builtin_amdgcn_tensor_load_to_lds(g0, g1, gz4, gz4, gz8, 0);
}

// ---------------------------------------------------------------------------
// K1: z[b,n,m] = sum_d |x[b,n,d] - y[b,m,d]| for a 32x32 tile per block.
// x/y tiles (32 rows x 128 f32 = 16KB each) DMA'd into LDS via TDM by wave 0.
// ---------------------------------------------------------------------------
__global__ __launch_bounds__(256)
void ssa_l1_tiles(const float* __restrict__ x, const float* __restrict__ y,
                  float* __restrict__ zbuf) {
    __shared__ __align__(16) float lx[32 * DD];
    __shared__ __align__(16) float ly[32 * DD];
    const int m0 = blockIdx.x * 32;
    const int n0 = blockIdx.y * 32;
    const int b  = blockIdx.z;

    if (threadIdx.x < 32) {  // wave 0 only (uniform branch under wave32)
        tdm_load_tile_f32(lds_offset(lx), x + ((size_t)b * NN + n0) * DD, DD, 32, DD);
        tdm_load_tile_f32(lds_offset(ly), y + ((size_t)b * MM + m0) * DD, DD, 32, DD);
        __builtin_amdgcn_s_wait_tensorcnt(0);
    }
    __syncthreads();

    const int l = threadIdx.x & 31;   // m index within tile (lane)
    const int w = threadIdx.x >> 5;   // wave id -> rows w*4 .. w*4+3
    float acc0 = 0.f, acc1 = 0.f, acc2 = 0.f, acc3 = 0.f;

    #pragma unroll 4
    for (int k = 0; k < 32; ++k) {
        // Lane-rotated d start spreads LDS banks (no padding needed for TDM).
        const int d = ((l + k) & 31) << 2;
        const float4 yv = *(const float4*)&ly[l * DD + d];
        const float4 x0 = *(const float4*)&lx[(w * 4 + 0) * DD + d];
        const float4 x1 = *(const float4*)&lx[(w * 4 + 1) * DD + d];
        const float4 x2 = *(const float4*)&lx[(w * 4 + 2) * DD + d];
        const float4 x3 = *(const float4*)&lx[(w * 4 + 3) * DD + d];
        acc0 += fabsf(x0.x - yv.x) + fabsf(x0.y - yv.y) + fabsf(x0.z - yv.z) + fabsf(x0.w - yv.w);
        acc1 += fabsf(x1.x - yv.x) + fabsf(x1.y - yv.y) + fabsf(x1.z - yv.z) + fabsf(x1.w - yv.w);
        acc2 += fabsf(x2.x - yv.x) + fabsf(x2.y - yv.y) + fabsf(x2.z - yv.z) + fabsf(x2.w - yv.w);
        acc3 += fabsf(x3.x - yv.x) + fabsf(x3.y - yv.y) + fabsf(x3.z - yv.z) + fabsf(x3.w - yv.w);
    }

    float* zt = zbuf + ((size_t)b * NN + n0) * MM + m0;
    zt[(w * 4 + 0) * MM + l] = acc0;
    zt[(w * 4 + 1) * MM + l] = acc1;
    zt[(w * 4 + 2) * MM + l] = acc2;
    zt[(w * 4 + 3) * MM + l] = acc3;
}

// ---------------------------------------------------------------------------
// K2: per-column (softmax over n) online max & sum-exp. Thread = one column m.
// Coalesced: lane l reads z[n, m0+l] (consecutive floats).
// ---------------------------------------------------------------------------
__global__ __launch_bounds__(256)
void ssa_col_stats(const float* __restrict__ zbuf, const int* __restrict__ xlen,
                   float* __restrict__ cmax, float* __restrict__ csum) {
    const int b  = blockIdx.y;
    const int m  = blockIdx.x * 256 + threadIdx.x;
    const int xl = xlen[b];
    const float* zb = zbuf + (size_t)b * NN * MM;
    float mx = -INFINITY, s = 0.f;
    for (int n = 0; n < xl; ++n) {
        const float v  = -zb[(size_t)n * MM + m];
        const float nm = fmaxf(mx, v);
        s  = s * expf(mx - nm) + expf(v - nm);
        mx = nm;
    }
    cmax[b * MM + m] = mx;
    csum[b * MM + m] = s;
}

// ---------------------------------------------------------------------------
// K3: per-row (softmax over m) online max & sum-exp. One wave32 per row,
// lanes stride the row, combine with wave32 xor-shuffles.
// ---------------------------------------------------------------------------
__global__ __launch_bounds__(256)
void ssa_row_stats(const float* __restrict__ zbuf, const int* __restrict__ ylen,
                   float* __restrict__ rmax, float* __restrict__ rsum) {
    const int b  = blockIdx.y;
    const int w  = threadIdx.x >> 5;
    const int l  = threadIdx.x & 31;
    const int n  = blockIdx.x * 8 + w;
    const int yl = ylen[b];
    const float* zr = zbuf + ((size_t)b * NN + n) * MM;
    float mx = -INFINITY, s = 0.f;
    for (int m = l; m < yl; m += 32) {
        const float v  = -zr[m];
        const float nm = fmaxf(mx, v);
        s  = s * expf(mx - nm) + expf(v - nm);
        mx = nm;
    }
    float mxa = mx;
    for (int off = 16; off; off >>= 1) mxa = fmaxf(mxa, __shfl_xor(mxa, off, 32));
    float sa = s * expf(mx - mxa);
    for (int off = 16; off; off >>= 1) sa += __shfl_xor(sa, off, 32);
    if (l == 0) { rmax[b * NN + n] = mxa; rsum[b * NN + n] = sa; }
}

// ---------------------------------------------------------------------------
// K4: per batch: S1 = sum a*z, S2 = sum a over valid region; out = -S1/S2.
// Deterministic fixed-order accumulation + LDS tree reduce.
// ---------------------------------------------------------------------------
__global__ __launch_bounds__(256)
void ssa_finalize(const float* __restrict__ zbuf,
                  const int* __restrict__ xlen, const int* __restrict__ ylen,
                  const float* __restrict__ cmax, const float* __restrict__ csum,
                  const float* __restrict__ rmax, const float* __restrict__ rsum,
                  float* __restrict__ out) {
    const int b  = blockIdx.x;
    const int t  = threadIdx.x;
    const int xl = xlen[b], yl = ylen[b];
    const float* zb = zbuf + (size_t)b * NN * MM;
    const float* cm = cmax + b * MM;
    const float* cs = csum + b * MM;
    float s1 = 0.f, s2 = 0.f;
    for (int n = 0; n < xl; ++n) {
        const float rm  = rmax[b * NN + n];
        const float rsv = rsum[b * NN + n];
        const float* zr = zb + (size_t)n * MM;
        for (int m = t; m < yl; m += 256) {
            const float zv = zr[m];
            const float wv = -zv;
            const float al = expf(wv - cm[m]) / cs[m];
            const float be = expf(wv - rm) / rsv;
            const float a  = al + be - al * be;
            s1 += a * zv;
            s2 += a;
        }
    }
    __shared__ float r1[256];
    __shared__ float r2[256];
    r1[t] = s1; r2[t] = s2;
    __syncthreads();
    for (int off = 128; off; off >>= 1) {
        if (t < off) { r1[t] += r1[t + off]; r2[t] += r2[t + off]; }
        __syncthreads();
    }
    if (t == 0) out[b] = -r1[0] / r2[0];
}

extern "C" void kernel_launch(void* const* d_in, const int* in_sizes, int n_in,
                              void* d_out, int out_size, void* d_ws, size_t ws_size,
                              hipStream_t stream) {
    (void)in_sizes; (void)n_in; (void)out_size; (void)ws_size;
    const float* x    = (const float*)d_in[0];
    const float* y    = (const float*)d_in[1];
    const int*   xlen = (const int*)d_in[2];   // int32 on device (jax x64 off)
    const int*   ylen = (const int*)d_in[3];
    float* out = (float*)d_out;

    // Workspace layout: z (4 MB) + column/row stats (32 KB).
    float* zbuf = (float*)d_ws;                           // BB*NN*MM
    float* cmax = zbuf + (size_t)BB * NN * MM;            // BB*MM
    float* csum = cmax + BB * MM;                         // BB*MM
    float* rmax = csum + BB * MM;                         // BB*NN
    float* rsum = rmax + BB * NN;                         // BB*NN

    ssa_l1_tiles <<<dim3(MM / 32, NN / 32, BB), 256, 0, stream>>>(x, y, zbuf);
    ssa_col_stats<<<dim3(MM / 256, BB),         256, 0, stream>>>(zbuf, xlen, cmax, csum);
    ssa_row_stats<<<dim3(NN / 8, BB),           256, 0, stream>>>(zbuf, ylen, rmax, rsum);
    ssa_finalize <<<dim3(BB),                   256, 0, stream>>>(zbuf, xlen, ylen,
                                                                 cmax, csum, rmax, rsum, out);
}